// TripletLoss_18202071400801
// MI455X (gfx1250) — compile-verified
//
#include <hip/hip_runtime.h>
#include <math.h>

// ---- problem constants (match reference) ----
#define D_DIM   512
#define B_TXT   4096
#define NIMG    10
#define N_DIST  (B_TXT * NIMG)        // 40960 distances: dist[g] = ||text[g/10] - image[g] + eps||
#define N_TRIP  (B_TXT * (NIMG - 1))  // 36864 triplets
#define MARGIN  1.0f
#define EPS_F   1e-6f

typedef __attribute__((ext_vector_type(2))) float v2f;
typedef __attribute__((ext_vector_type(8))) float v8f;

// ---------------------------------------------------------------------------
// Kernel 1: per-wave, compute 16 Euclidean distances (each over 512 dims).
//   lane = (m, h): m = lane&15 selects row g = base+m, h = lane>>4 selects the
//   dim half [h*256, h*256+256). acc0 covers the first 128 dims of the half,
//   acc1 the second 128 -> these are exactly A[m][2h] and A[m][2h+1] in the
//   CDNA5 16x4 f32 A-matrix VGPR layout. One v_wmma_f32_16x16x4_f32 with an
//   all-ones B matrix produces D[m][*] = sum of row m's 4 partials, i.e. the
//   squared norm, computed entirely by the matrix pipe (no shuffle tree).
// ---------------------------------------------------------------------------
__global__ void __launch_bounds__(256)
triplet_dist_kernel(const float* __restrict__ text,
                    const float* __restrict__ image,
                    float* __restrict__ dist)
{
    const int tid  = blockIdx.x * blockDim.x + threadIdx.x;
    const int wave = tid >> 5;            // global wave id, wave32
    const int lane = threadIdx.x & 31;
    const int base = wave * 16;           // first distance id of this wave's tile

    const int m = lane & 15;              // row within tile
    const int h = lane >> 4;              // which 256-dim half
    const int g = base + m;               // distance id == image row id
    const int a = (int)((unsigned)g / 10u);  // anchor/text row id

    const float4* __restrict__ trow =
        (const float4*)(text  + (size_t)a * D_DIM + (size_t)h * 256);
    const float4* __restrict__ irow =
        (const float4*)(image + (size_t)g * D_DIM + (size_t)h * 256);

    float acc0 = 0.0f;  // dims [h*256      , h*256 + 128)
    float acc1 = 0.0f;  // dims [h*256 + 128, h*256 + 256)

#pragma unroll 8
    for (int it = 0; it < 32; ++it) {
        float4 t  = trow[it];
        float4 im = irow[it];
        float d0 = t.x - im.x + EPS_F;
        float d1 = t.y - im.y + EPS_F;
        float d2 = t.z - im.z + EPS_F;
        float d3 = t.w - im.w + EPS_F;
        acc0 = fmaf(d0, d0, acc0);
        acc0 = fmaf(d1, d1, acc0);
        acc0 = fmaf(d2, d2, acc0);
        acc0 = fmaf(d3, d3, acc0);
    }
#pragma unroll 8
    for (int it = 32; it < 64; ++it) {
        float4 t  = trow[it];
        float4 im = irow[it];
        float d0 = t.x - im.x + EPS_F;
        float d1 = t.y - im.y + EPS_F;
        float d2 = t.z - im.z + EPS_F;
        float d3 = t.w - im.w + EPS_F;
        acc1 = fmaf(d0, d0, acc1);
        acc1 = fmaf(d1, d1, acc1);
        acc1 = fmaf(d2, d2, acc1);
        acc1 = fmaf(d3, d3, acc1);
    }

    // A: 16x4 partial-sum matrix (2 f32 per lane).  B: all-ones 4x16 (layout-
    // agnostic since every element is 1).  D[m][n] = sum_k A[m][k] = ||.||^2.
    v2f A; A.x = acc0; A.y = acc1;
    v2f Bm; Bm.x = 1.0f; Bm.y = 1.0f;
    v8f C = {};
    v8f Dv = __builtin_amdgcn_wmma_f32_16x16x4_f32(
        /*neg_a=*/false, A, /*neg_b=*/false, Bm,
        /*c_mod=*/(short)0, C, /*reuse_a=*/false, /*reuse_b=*/false);

    // C/D layout: VGPR r holds row r (lanes 0-15) and row 8+r (lanes 16-31).
    // Lanes 0-7 write rows 0-7; lanes 16-23 write rows 8-15.
    const int  r      = lane & 7;
    const bool lowWr  = (lane < 8);
    const bool hiWr   = (lane >= 16) && (lane < 24);
    if (lowWr | hiWr) {
        float sq  = Dv[r];                       // dynamic extract from v8f
        int   row = lowWr ? r : (8 + r);
        dist[base + row] = sqrtf(sq);
    }
}

// ---------------------------------------------------------------------------
// Kernel 2: hinge + mean over 36864 triplets. Single block, deterministic
// LDS tree reduction (no float atomics -> bit-stable across graph replays).
// ---------------------------------------------------------------------------
__global__ void __launch_bounds__(1024)
triplet_loss_reduce_kernel(const float* __restrict__ dist,
                           const int*  __restrict__ pos_idx,
                           const int*  __restrict__ neg_idx,
                           float* __restrict__ out)
{
    __shared__ float red[1024];
    const int tid = threadIdx.x;

    float s = 0.0f;
    for (int t = tid; t < N_TRIP; t += 1024) {
        float d_ap = dist[pos_idx[t]];   // pos = a*10   -> dist pairs text[a]
        float d_an = dist[neg_idx[t]];   // neg = a*10+j -> same anchor a
        float l = d_ap - d_an + MARGIN;
        s += (l > 0.0f) ? l : 0.0f;
    }
    red[tid] = s;
    __syncthreads();

#pragma unroll
    for (int off = 512; off > 0; off >>= 1) {
        if (tid < off) red[tid] += red[tid + off];
        __syncthreads();
    }
    if (tid == 0) out[0] = red[0] * (1.0f / (float)N_TRIP);
}

// ---------------------------------------------------------------------------
extern "C" void kernel_launch(void* const* d_in, const int* in_sizes, int n_in,
                              void* d_out, int out_size, void* d_ws, size_t ws_size,
                              hipStream_t stream)
{
    (void)in_sizes; (void)n_in; (void)out_size; (void)ws_size;

    const float* text  = (const float*)d_in[0];   // [4096, 512] f32
    const float* image = (const float*)d_in[1];   // [40960, 512] f32
    // d_in[2] = anchor_idx (unused: implied by pos/neg structure)
    const int* pos_idx = (const int*)d_in[3];     // [36864] i32
    const int* neg_idx = (const int*)d_in[4];     // [36864] i32
    float* out  = (float*)d_out;
    float* dist = (float*)d_ws;                   // 40960 f32 = 160 KB scratch

    // 40960 distances / 16 per wave = 2560 waves = 320 blocks x 256 threads.
    triplet_dist_kernel<<<dim3(N_DIST / 16 / 8), dim3(256), 0, stream>>>(
        text, image, dist);

    triplet_loss_reduce_kernel<<<dim3(1), dim3(1024), 0, stream>>>(
        dist, pos_idx, neg_idx, out);
}